// hash_embedding_68109591380122
// MI455X (gfx1250) — compile-verified
//
#include <hip/hip_runtime.h>
#include <stdint.h>
#include <math.h>

// ---------------------------------------------------------------------------
// Hash-grid encode (Instant-NGP style), MI455X / gfx1250, wave32.
// Bottleneck: 134M random 8B gathers from 64MB of tables (L2-resident on the
// 192MB L2). Levers: NT hints on the 136MB streaming output, deep gather MLP,
// async-DMA staging of x, and minimized per-thread VALU (lane-parallel level
// resolutions + rcp instead of 6 divides/level).
// ---------------------------------------------------------------------------

typedef __attribute__((ext_vector_type(2))) float v2f;
typedef __attribute__((ext_vector_type(4))) float v4f;
typedef __attribute__((ext_vector_type(4))) int   v4i;

#define GLOBAL_AS __attribute__((address_space(1)))
#define LDS_AS    __attribute__((address_space(3)))

namespace {
constexpr int      kLevels = 16;
constexpr unsigned kTSize  = 1u << 19;       // 524288 entries / level
constexpr unsigned kHMask  = kTSize - 1u;
constexpr int      kBlock  = 256;            // 8 wave32s
}

#if defined(__HIP_DEVICE_COMPILE__) && __has_builtin(__builtin_amdgcn_global_load_async_to_lds_b128)
#define USE_ASYNC_LDS 1
#else
#define USE_ASYNC_LDS 0
#endif

// One hash-grid level: 8 gathered float2 corners, trilinear blend.
// invN = 1/Nl, invsp* = 1/span (hoisted): no divides inside.
__device__ __forceinline__ v2f enc_level(const v2f* __restrict__ tab, float Nl,
                                         float invN,
                                         float px, float py, float pz,
                                         float ux, float uy, float uz,
                                         float mnx, float mny, float mnz,
                                         float spx, float spy, float spz,
                                         float invspx, float invspy, float invspz) {
  // Grid cell: identical arithmetic to the reference (floor boundaries safe).
  const float fx = floorf(ux * Nl);
  const float fy = floorf(uy * Nl);
  const float fz = floorf(uz * Nl);
  // box_size = span/Nl (1 ulp vs divide); weights use the UNclipped x.
  const float bsx = spx * invN, bsy = spy * invN, bsz = spz * invN;
  const float ibx = Nl * invspx, iby = Nl * invspy, ibz = Nl * invspz;
  const float wx = (px - (fx * bsx + mnx)) * ibx;
  const float wy = (py - (fy * bsy + mny)) * iby;
  const float wz = (pz - (fz * bsz + mnz)) * ibz;

  const unsigned gx = (unsigned)(int)fx;
  const unsigned gy = (unsigned)(int)fy;
  const unsigned gz = (unsigned)(int)fz;
  // (g+1)*P == g*P + P (mod 2^32): 2 mul + 3 add for all 8 corners.
  const unsigned hx0 = gx;                   const unsigned hx1 = gx + 1u;
  const unsigned hy0 = gy * 2654435761u;     const unsigned hy1 = hy0 + 2654435761u;
  const unsigned hz0 = gz * 805459861u;      const unsigned hz1 = hz0 + 805459861u;

  // 8 independent b64 gathers -> all issued before first use (MLP).
  const v2f e000 = tab[(hx0 ^ hy0 ^ hz0) & kHMask];
  const v2f e001 = tab[(hx0 ^ hy0 ^ hz1) & kHMask];
  const v2f e010 = tab[(hx0 ^ hy1 ^ hz0) & kHMask];
  const v2f e011 = tab[(hx0 ^ hy1 ^ hz1) & kHMask];
  const v2f e100 = tab[(hx1 ^ hy0 ^ hz0) & kHMask];
  const v2f e101 = tab[(hx1 ^ hy0 ^ hz1) & kHMask];
  const v2f e110 = tab[(hx1 ^ hy1 ^ hz0) & kHMask];
  const v2f e111 = tab[(hx1 ^ hy1 ^ hz1) & kHMask];

  const float ox = 1.0f - wx, oy = 1.0f - wy, oz = 1.0f - wz;
  const v2f c00 = e000 * ox + e100 * wx;
  const v2f c01 = e001 * ox + e101 * wx;
  const v2f c10 = e010 * ox + e110 * wx;
  const v2f c11 = e011 * ox + e111 * wx;
  const v2f c0  = c00 * oy + c10 * wy;
  const v2f c1  = c01 * oy + c11 * wy;
  return c0 * oz + c1 * wz;
}

__global__ __launch_bounds__(kBlock) void hashgrid_encode_kernel(
    const float* __restrict__ x,       // (N,3)
    const float* __restrict__ bbox,    // (2,3): min row then max row
    const float* __restrict__ tables,  // (16, 2^19, 2)
    float* __restrict__ out,           // N*32 features, then N mask floats
    int n) {
  __shared__ __align__(16) float xs[kBlock * 3];

  const int tid  = (int)threadIdx.x;
  const int p0   = (int)blockIdx.x * kBlock;
  const int p    = p0 + tid;
  const int lane = tid & 31;
  const int wave = tid >> 5;
  (void)xs;

  // ---- Per-level resolutions, computed lane-parallel with ONE powf, then
  // broadcast via v_readlane (uniform SGPR values). Same f32 formula as the
  // reference so floor() boundaries match. Done while EXEC is still full.
  const float bg = expf((logf(512.0f) - logf(16.0f)) * (1.0f / 15.0f));
  const float nlLane = floorf(16.0f * powf(bg, (float)(lane & 15)));
  float Nl[kLevels];
#pragma unroll
  for (int l = 0; l < kLevels; ++l) {
    Nl[l] = __int_as_float(__builtin_amdgcn_readlane(__float_as_int(nlLane), l));
  }

  float px, py, pz;
  bool loaded = false;
#if USE_ASYNC_LDS
  if (p0 + kBlock <= n) {
    // Each wave DMAs its own 32 points (384B) of x into LDS as 24 b128s:
    // coalesced async path (ASYNCcnt), then per-lane LDS reads. Only this
    // wave reads this region, so s_wait_asynccnt alone orders it (no barrier).
    const float* gsrc = x + (size_t)(p0 + wave * 32) * 3;
    if (lane < 24) {
      __builtin_amdgcn_global_load_async_to_lds_b128(
          (GLOBAL_AS v4i*)(gsrc + lane * 4),
          (LDS_AS v4i*)(xs + wave * 96 + lane * 4),
          /*offset=*/0, /*cpol=*/0);
    }
#if __has_builtin(__builtin_amdgcn_s_wait_asynccnt)
    __builtin_amdgcn_s_wait_asynccnt(0);
#else
    asm volatile("s_wait_asynccnt 0" ::: "memory");
#endif
    px = xs[tid * 3 + 0];
    py = xs[tid * 3 + 1];
    pz = xs[tid * 3 + 2];
    loaded = true;
  }
#endif
  if (!loaded) {
    if (p >= n) return;
    px = x[(size_t)p * 3 + 0];
    py = x[(size_t)p * 3 + 1];
    pz = x[(size_t)p * 3 + 2];
  }
  if (p >= n) return;

  const float mnx = bbox[0], mny = bbox[1], mnz = bbox[2];
  const float mxx = bbox[3], mxy = bbox[4], mxz = bbox[5];
  const float spx = mxx - mnx, spy = mxy - mny, spz = mxz - mnz;
  const float invspx = 1.0f / spx, invspy = 1.0f / spy, invspz = 1.0f / spz;

  const bool keep = (px >= mnx) && (px <= mxx) &&
                    (py >= mny) && (py <= mxy) &&
                    (pz >= mnz) && (pz <= mxz);

  const float cx = fminf(fmaxf(px, mnx), mxx);
  const float cy = fminf(fmaxf(py, mny), mxy);
  const float cz = fminf(fmaxf(pz, mnz), mxz);
  const float ux = (cx - mnx) * invspx;
  const float uy = (cy - mny) * invspy;
  const float uz = (cz - mnz) * invspz;

  float* orow = out + (size_t)p * (kLevels * 2);
#pragma unroll
  for (int l = 0; l < kLevels; l += 2) {
    const float Nl0 = Nl[l],   invN0 = 1.0f / Nl0;
    const float Nl1 = Nl[l+1], invN1 = 1.0f / Nl1;
    const v2f* t0 = (const v2f*)tables + (size_t)l * kTSize;
    const v2f* t1 = t0 + kTSize;
    const v2f r0 = enc_level(t0, Nl0, invN0, px, py, pz, ux, uy, uz,
                             mnx, mny, mnz, spx, spy, spz, invspx, invspy, invspz);
    const v2f r1 = enc_level(t1, Nl1, invN1, px, py, pz, ux, uy, uz,
                             mnx, mny, mnz, spx, spy, spz, invspx, invspy, invspz);
    const v4f r = {r0.x, r0.y, r1.x, r1.y};
    // NT store: 136MB streaming output must not evict the 64MB L2-resident
    // tables (192MB L2) — the key cache-policy lever on MI455X.
    __builtin_nontemporal_store(r, (v4f*)orow + (l >> 1));
  }

  __builtin_nontemporal_store(keep ? 1.0f : 0.0f,
                              out + (size_t)n * (kLevels * 2) + p);
}

extern "C" void kernel_launch(void* const* d_in, const int* in_sizes, int n_in,
                              void* d_out, int out_size, void* d_ws, size_t ws_size,
                              hipStream_t stream) {
  (void)n_in; (void)out_size; (void)d_ws; (void)ws_size;
  const float* x      = (const float*)d_in[0];
  const float* bbox   = (const float*)d_in[1];
  const float* tables = (const float*)d_in[2];
  float* out          = (float*)d_out;
  const int n = in_sizes[0] / 3;
  const int blocks = (n + kBlock - 1) / kBlock;
  hashgrid_encode_kernel<<<blocks, kBlock, 0, stream>>>(x, bbox, tables, out, n);
}